// PoincareAttention_63642825392412
// MI455X (gfx1250) — compile-verified
//
#include <hip/hip_runtime.h>
#include <hip/hip_bf16.h>
#include <math.h>

typedef __attribute__((ext_vector_type(2))) float v2f;
typedef __attribute__((ext_vector_type(8))) float v8f;

#define T_SEQ   2048
#define D_DIM   64
#define NBATCH  8
#define BM      64            // query rows per block
#define WAVES   4             // 1 wave -> 16 query rows
#define KTILE   32            // keys per iteration (2 x 16-wide WMMA subtiles)
#define NKT     (T_SEQ / KTILE)

#define HALF_LN2   0.34657359027997264f   // 0.5*ln2 (for arctanh via log2)
#define TWO_OV_LN2 2.8853900817779268f    // 2/ln2  (for tanh via exp2)

__device__ __forceinline__ float rcp_f32(float x)  { return __builtin_amdgcn_rcpf(x); }
__device__ __forceinline__ float sqrt_f32(float x) { return __builtin_amdgcn_sqrtf(x); }
__device__ __forceinline__ float exp2_f32(float x) { return __builtin_amdgcn_exp2f(x); }
__device__ __forceinline__ float log2_f32(float x) { return __builtin_amdgcn_logf(x); }

__device__ __forceinline__ float group16_max(float v) {
    v = fmaxf(v, __shfl_xor(v, 1, 32));
    v = fmaxf(v, __shfl_xor(v, 2, 32));
    v = fmaxf(v, __shfl_xor(v, 4, 32));
    v = fmaxf(v, __shfl_xor(v, 8, 32));
    return v;
}
__device__ __forceinline__ float group16_sum(float v) {
    v += __shfl_xor(v, 1, 32);
    v += __shfl_xor(v, 2, 32);
    v += __shfl_xor(v, 4, 32);
    v += __shfl_xor(v, 8, 32);
    return v;
}

// Prologue: per row of x (8*2048 rows, 64 wide) compute
//   xs[row]          = sum(x^2)            (raw, clipping applied later)
//   vtT[n][d][t]     = transposed v_tan:  x * arctanh(nc)/nc
// The transpose makes the PV-stage WMMA B fragments ({V[k][d],V[k+1][d]})
// contiguous float2 loads in the attention kernel.
__global__ void poincare_prep(const float* __restrict__ x,
                              float* __restrict__ xs,
                              float* __restrict__ vtT) {
    const int wave = threadIdx.x >> 5;
    const int lane = threadIdx.x & 31;
    const int row  = blockIdx.x * 8 + wave;          // 0..16383
    const size_t base = (size_t)row * D_DIM + lane * 2;
    v2f xv = *(const v2f*)(x + base);
    float ss = xv[0] * xv[0] + xv[1] * xv[1];
    ss += __shfl_xor(ss, 1, 32);
    ss += __shfl_xor(ss, 2, 32);
    ss += __shfl_xor(ss, 4, 32);
    ss += __shfl_xor(ss, 8, 32);
    ss += __shfl_xor(ss, 16, 32);
    if (lane == 0) xs[row] = ss;
    float n  = sqrt_f32(ss);
    float nc = fminf(fmaxf(n, 1e-15f), 0.99999f);
    // arctanh(nc)/nc = 0.5*ln2*log2((1+nc)/(1-nc)) / nc
    float scale = HALF_LN2 * log2_f32((1.0f + nc) * rcp_f32(1.0f - nc)) * rcp_f32(nc);
    const int nb = row >> 11;                // batch 0..7
    const int t  = row & (T_SEQ - 1);
    float* vc = vtT + ((size_t)nb * D_DIM + lane * 2) * T_SEQ + t;
    vc[0]     = xv[0] * scale;               // d = 2*lane
    vc[T_SEQ] = xv[1] * scale;               // d = 2*lane+1
}

// Flash-style Poincare attention. One wave = 16 query rows of one batch.
__global__ void __launch_bounds__(WAVES * 32)
poincare_flash(const float* __restrict__ x,     // (8,2048,64)
               const float* __restrict__ xs,    // (8,2048)
               const float* __restrict__ vtT,   // (8,64,2048)  d-major v_tan
               const float* __restrict__ temp,  // (1,)
               const float* __restrict__ Wout,  // (64,64) row-major
               const float* __restrict__ bout,  // (64,)
               float* __restrict__ out)         // (8,2048,64) flat == (B,T,P,D)
{
    __shared__ float lds_p[WAVES][16 * 36];   // P tile (16x32) transpose staging
    __shared__ float lds_o[WAVES][16 * 66];   // O tile transpose staging

    const int wave = threadIdx.x >> 5;
    const int lane = threadIdx.x & 31;
    const int l15  = lane & 15;
    const int hi   = lane >> 4;               // which half of the wave
    const int blk  = blockIdx.x;              // 0..255
    const int n    = blk >> 5;                // batch 0..7
    const int qbase = (blk & 31) * BM + wave * 16;

    const float* xb  = x   + (size_t)n * T_SEQ * D_DIM;
    const float* vb  = vtT + (size_t)n * D_DIM * T_SEQ;
    const float* xsb = xs + n * T_SEQ;

    const float invT = rcp_f32(fmaxf(temp[0], 0.01f));   // (also log2-domain scale)

    // Preload Q A-fragments: 16 chunks of K=4 (f32 WMMA A layout:
    // lane<16 -> {K0,K1}, lane>=16 -> {K2,K3}, M = lane&15)
    v2f a[16];
    {
        const float* qr = xb + (size_t)(qbase + l15) * D_DIM + hi * 2;
        #pragma unroll
        for (int kc = 0; kc < 16; ++kc)
            a[kc] = *(const v2f*)(qr + kc * 4);
    }
    // Per-query-row stats (component i covers row i (lanes 0-15) / i+8 (16-31))
    float xsq[8], alq[8];
    #pragma unroll
    for (int i = 0; i < 8; ++i) {
        float v = xsb[qbase + i + 8 * hi];
        xsq[i] = v;
        alq[i] = 1.0f - fminf(v, 0.99f);
    }

    float m8[8], l8[8];     // online-softmax state, log2 domain
    v8f o[4];
    #pragma unroll
    for (int i = 0; i < 8; ++i) { m8[i] = -INFINITY; l8[i] = 0.0f; }
    #pragma unroll
    for (int nt = 0; nt < 4; ++nt) o[nt] = (v8f){};

    float* pw = lds_p[wave];
    float* ow = lds_o[wave];

    for (int j = 0; j < NKT; ++j) {
        const int kbase = j * KTILE;
        if (j + 1 < NKT) {   // pull next key tile toward the WGP
            __builtin_prefetch(xb + (size_t)(kbase + KTILE + l15) * D_DIM, 0, 1);
            __builtin_prefetch(vb + (size_t)(lane * 2) * T_SEQ + kbase + KTILE, 0, 1);
        }

        // ---- Two 16x16 score subtiles: S = Q(16x64) @ K(16x64)^T
        v8f c0 = (v8f){}, c1 = (v8f){};
        const float* kr0 = xb + (size_t)(kbase + l15) * D_DIM + hi * 2;
        const float* kr1 = kr0 + 16 * D_DIM;
        #pragma unroll
        for (int kc = 0; kc < 16; ++kc) {
            v2f bf = *(const v2f*)(kr0 + kc * 4);
            c0 = __builtin_amdgcn_wmma_f32_16x16x4_f32(false, a[kc], false, bf,
                                                       (short)0, c0, false, false);
        }
        #pragma unroll
        for (int kc = 0; kc < 16; ++kc) {
            v2f bf = *(const v2f*)(kr1 + kc * 4);
            c1 = __builtin_amdgcn_wmma_f32_16x16x4_f32(false, a[kc], false, bf,
                                                       (short)0, c1, false, false);
        }

        const float xsk0 = xsb[kbase + l15];
        const float xsk1 = xsb[kbase + 16 + l15];
        const float bek0 = 1.0f - fminf(xsk0, 0.99f);
        const float bek1 = 1.0f - fminf(xsk1, 0.99f);

        // ---- Poincare distance + online softmax (log2 domain, shared update)
        #pragma unroll
        for (int i = 0; i < 8; ++i) {
            float rab0 = rcp_f32(alq[i] * bek0 + 1e-8f);
            float rab1 = rcp_f32(alq[i] * bek1 + 1e-8f);
            float d20  = fmaxf(xsq[i] + xsk0 - 2.0f * c0[i], 0.0f);
            float d21  = fmaxf(xsq[i] + xsk1 - 2.0f * c1[i], 0.0f);
            float dl0  = fmaxf(1.0f + 2.0f * d20 * rab0, 1.0f + 1e-6f);
            float dl1  = fmaxf(1.0f + 2.0f * d21 * rab1, 1.0f + 1e-6f);
            // L = -log2(delta + sqrt(delta^2-1)) * invT   (arccosh, base-2)
            float L0 = -log2_f32(dl0 + sqrt_f32(dl0 * dl0 - 1.0f)) * invT;
            float L1 = -log2_f32(dl1 + sqrt_f32(dl1 * dl1 - 1.0f)) * invT;

            float tm   = group16_max(fmaxf(L0, L1));
            float mnew = fmaxf(m8[i], tm);
            float corr = exp2_f32(m8[i] - mnew);
            float p0   = exp2_f32(L0 - mnew);
            float p1   = exp2_f32(L1 - mnew);
            float rs   = group16_sum(p0 + p1);
            l8[i] = l8[i] * corr + rs;
            m8[i] = mnew;
            #pragma unroll
            for (int nt = 0; nt < 4; ++nt) o[nt][i] *= corr;
            pw[(i + 8 * hi) * 36 + l15]      = p0;   // stage P for A-layout read
            pw[(i + 8 * hi) * 36 + 16 + l15] = p1;
        }
        __syncthreads();

        // ---- O += P(16x32) @ V(32x64), P re-read in A layout from LDS,
        //      V B-fragments are contiguous float2 in the transposed layout
        v2f pa[8];
        #pragma unroll
        for (int kc = 0; kc < 8; ++kc)
            pa[kc] = *(const v2f*)(pw + l15 * 36 + kc * 4 + hi * 2);
        #pragma unroll
        for (int kc = 0; kc < 8; ++kc) {
            #pragma unroll
            for (int nt = 0; nt < 4; ++nt) {
                v2f bf = *(const v2f*)(vb + (size_t)(nt * 16 + l15) * T_SEQ
                                          + kbase + kc * 4 + hi * 2);
                o[nt] = __builtin_amdgcn_wmma_f32_16x16x4_f32(false, pa[kc], false, bf,
                                                              (short)0, o[nt], false, false);
            }
        }
        __syncthreads();
    }

    // ---- finalize softmax + exp_map_zero, stage O for the projection GEMM
    #pragma unroll
    for (int i = 0; i < 8; ++i) {
        float invl = rcp_f32(l8[i]);          // l >= 1 always
        float ss = 0.0f;
        #pragma unroll
        for (int nt = 0; nt < 4; ++nt) { o[nt][i] *= invl; ss += o[nt][i] * o[nt][i]; }
        ss = group16_sum(ss);
        float nn = sqrt_f32(ss);
        float nc = fmaxf(nn, 1e-15f);
        // tanh(nc) = 1 - 2/(exp2(nc*2/ln2)+1), overflow-safe
        float th = 1.0f - 2.0f * rcp_f32(exp2_f32(nc * TWO_OV_LN2) + 1.0f);
        float sc = th * rcp_f32(nc);
        #pragma unroll
        for (int nt = 0; nt < 4; ++nt) {
            o[nt][i] *= sc;
            ow[(i + 8 * hi) * 66 + nt * 16 + l15] = o[nt][i];
        }
    }
    __syncthreads();

    // ---- r = out(16x64) @ Wout^T(64x64) + b  (64 WMMA ops)
    v8f r[4];
    #pragma unroll
    for (int nt = 0; nt < 4; ++nt) {
        float bv = bout[nt * 16 + l15];
        #pragma unroll
        for (int i = 0; i < 8; ++i) r[nt][i] = bv;
    }
    #pragma unroll
    for (int kc = 0; kc < 16; ++kc) {
        v2f pa = *(const v2f*)(ow + l15 * 66 + kc * 4 + hi * 2);
        #pragma unroll
        for (int nt = 0; nt < 4; ++nt) {
            // B[k][nout] = Wout[nout][k] -> consecutive float2 along each Wout row
            v2f wb = *(const v2f*)(Wout + (size_t)(nt * 16 + l15) * D_DIM + kc * 4 + hi * 2);
            r[nt] = __builtin_amdgcn_wmma_f32_16x16x4_f32(false, pa, false, wb,
                                                          (short)0, r[nt], false, false);
        }
    }

    // ---- store: output flat layout matches (8, T, 64)
    float* ob = out + (size_t)n * T_SEQ * D_DIM;
    #pragma unroll
    for (int nt = 0; nt < 4; ++nt)
        #pragma unroll
        for (int i = 0; i < 8; ++i)
            ob[(size_t)(qbase + i + 8 * hi) * D_DIM + nt * 16 + l15] = r[nt][i];
}

extern "C" void kernel_launch(void* const* d_in, const int* in_sizes, int n_in,
                              void* d_out, int out_size, void* d_ws, size_t ws_size,
                              hipStream_t stream) {
    const float* coords      = (const float*)d_in[0];
    // d_in[1] = values: unused by the reference forward
    const float* temperature = (const float*)d_in[2];
    const float* W_out       = (const float*)d_in[3];
    const float* b_out       = (const float*)d_in[4];
    float* out = (float*)d_out;

    float* xs  = (float*)d_ws;                        // 8*2048 floats
    float* vtT = xs + NBATCH * T_SEQ;                 // 8*64*2048 floats (~4 MB)

    poincare_prep<<<dim3((NBATCH * T_SEQ) / 8), dim3(256), 0, stream>>>(coords, xs, vtT);
    poincare_flash<<<dim3(NBATCH * (T_SEQ / BM)), dim3(WAVES * 32), 0, stream>>>(
        coords, xs, vtT, temperature, W_out, b_out, out);
}